// BinDensityEncoder_60258391163074
// MI455X (gfx1250) — compile-verified
//
#include <hip/hip_runtime.h>
#include <stdint.h>

// BinDensityEncoder: per-row 3D histogram (32^3 = 32768 bins), counts / N.
// states: [512, 8192, 3] f32   edges: linspace(-3,3,33)   out: [512, 32768] f32
//
// Strategy (MI455X / gfx1250):
//  - 1 workgroup (1024 threads = 32 waves) per batch row.
//  - Private 128 KB u32 histogram in LDS (CDNA5 WGP has 320 KB LDS) -> all
//    scatter-adds are ds_add_u32; global memory touched exactly once in, once out.
//  - Input row (96 KB) staged into LDS via CDNA5 async-to-LDS copies
//    (global_load_async_to_lds_b128, ASYNCcnt), double-buffered in two 48 KB
//    chunks. Each thread copies and later consumes only its own 48 B slice, so
//    only a per-wave s_wait_asynccnt is needed (no extra barriers); the copies
//    overlap with zeroing the histogram.

namespace {
constexpr int kBins        = 32;
constexpr int kS           = 3;
constexpr int kB           = 512;
constexpr int kN           = 8192;
constexpr int kOutDim      = kBins * kBins * kBins;          // 32768
constexpr int kThreads     = 1024;                           // 32 waves (wave32)
constexpr int kChunkSamp   = 4096;                           // samples per chunk
constexpr int kChunkBytes  = kChunkSamp * kS * 4;            // 49152 B
constexpr int kHistBytes   = kOutDim * 4;                    // 131072 B
constexpr int kSmemBytes   = kHistBytes + 2 * kChunkBytes;   // 229376 B <= 320 KB
constexpr int kSampPerThr  = kChunkSamp / kThreads;          // 4 samples = 48 B/thread
}

// One 16-byte async global->LDS copy (per-lane addresses). Tracked by ASYNCcnt.
__device__ __forceinline__ void async_ld16(uint32_t lds_byte_addr, const void* gptr) {
  asm volatile("global_load_async_to_lds_b128 %0, %1, off"
               :
               : "v"(lds_byte_addr), "v"((uint64_t)(uintptr_t)gptr)
               : "memory");
}

// torch.bucketize(x, edges[1:-1], right=False) over a uniform grid.
// Edges are exact in fp32 (multiples of 3*2^-4), so compute arithmetically and
// apply a one-step strict-less boundary correction.
__device__ __forceinline__ int bin_of(float x) {
  const float lo = -3.0f;
  const float hi = 3.0f - 1e-8f;            // folds to 3.0f, same as fp32 JAX
  x = fminf(fmaxf(x, lo), hi);
  float t  = (x - lo) * (32.0f / 6.0f);     // (x - lo) is exact; scale is ~1ulp
  int idx  = (int)t;                        // t >= 0 -> trunc == floor
  idx = idx > (kBins - 1) ? (kBins - 1) : idx;
  float edge_lo = fmaf((float)idx, 0.1875f, -3.0f);   // exact
  float edge_hi = edge_lo + 0.1875f;                  // exact
  if (idx < kBins - 1 && x > edge_hi)      idx++;     // scale rounded us low
  else if (idx > 0 && x <= edge_lo)        idx--;     // value on/below lower edge
  return idx;
}

__global__ __launch_bounds__(kThreads)
void BinDensityEncoder_hist_kernel(const float* __restrict__ states,
                                   float* __restrict__ out) {
  extern __shared__ char smem[];
  unsigned int* hist = reinterpret_cast<unsigned int*>(smem);

  const int tid = threadIdx.x;
  const int b   = blockIdx.x;
  const float* row = states + (size_t)b * kN * kS;

  // Generic shared pointer low 32 bits == LDS byte address (aperture form).
  const uint32_t smem_base = (uint32_t)(uintptr_t)smem;

  // Kick off async staging of both 48 KB input chunks. Thread t owns the
  // 48-byte slice [t*48, t*48+48) of each chunk (its 4 samples).
  #pragma unroll
  for (int c = 0; c < 2; ++c) {
    const uint32_t l = smem_base + kHistBytes + c * kChunkBytes + tid * 48;
    const char*    g = (const char*)row + c * kChunkBytes + tid * 48;
    async_ld16(l +  0, g +  0);
    async_ld16(l + 16, g + 16);
    async_ld16(l + 32, g + 32);
  }

  // Zero the 32768-bin histogram while async copies are in flight.
  for (int i = tid; i < kOutDim; i += kThreads) hist[i] = 0u;
  __syncthreads();

  // Consume chunks. Async loads complete in order per wave: after issuing 6
  // b128 copies, asynccnt<=3 means chunk 0 is resident, <=0 means chunk 1 is.
  #pragma unroll
  for (int c = 0; c < 2; ++c) {
    if (c == 0) asm volatile("s_wait_asynccnt 0x3" ::: "memory");
    else        asm volatile("s_wait_asynccnt 0x0" ::: "memory");
    const float* stage =
        reinterpret_cast<const float*>(smem + kHistBytes + c * kChunkBytes + tid * 48);
    #pragma unroll
    for (int j = 0; j < kSampPerThr; ++j) {
      const float x0 = stage[j * 3 + 0];
      const float x1 = stage[j * 3 + 1];
      const float x2 = stage[j * 3 + 2];
      const int lin = bin_of(x0) | (bin_of(x1) << 5) | (bin_of(x2) << 10);
      atomicAdd(&hist[lin], 1u);            // ds_add_u32 (no return)
    }
  }
  __syncthreads();

  // Emit counts / N, coalesced float4 stores (each wave writes 512 contiguous B).
  const float inv_n = 1.0f / (float)kN;
  float* orow = out + (size_t)b * kOutDim;
  #pragma unroll
  for (int i = 0; i < kOutDim / (kThreads * 4); ++i) {   // 8 iterations
    const int base = tid * 4 + i * kThreads * 4;
    const uint4 c4 = *reinterpret_cast<const uint4*>(hist + base);
    float4 v;
    v.x = (float)c4.x * inv_n;
    v.y = (float)c4.y * inv_n;
    v.z = (float)c4.z * inv_n;
    v.w = (float)c4.w * inv_n;
    *reinterpret_cast<float4*>(orow + base) = v;
  }
}

extern "C" void kernel_launch(void* const* d_in, const int* in_sizes, int n_in,
                              void* d_out, int out_size, void* d_ws, size_t ws_size,
                              hipStream_t stream) {
  (void)in_sizes; (void)n_in; (void)d_ws; (void)ws_size; (void)out_size;
  const float* states = (const float*)d_in[0];   // [512, 8192, 3] f32
  // d_in[1] = edges (exact uniform linspace, reproduced arithmetically)
  // d_in[2] = bins (compile-time constant 32)
  float* out = (float*)d_out;                    // [512, 32768] f32

  dim3 grid(kB), block(kThreads);
  hipLaunchKernelGGL(BinDensityEncoder_hist_kernel, grid, block, kSmemBytes,
                     stream, states, out);
}